// L_correspondence_15221364097727
// MI455X (gfx1250) — compile-verified
//
#include <hip/hip_runtime.h>
#include <hip/hip_bf16.h>
#include <math.h>

// Problem constants (match reference: H=W=256, SMALL=8, LARGE=16, B=2)
#define WIN    1024      // (256/8)^2 windows
#define NB     2         // batch
#define S2     64        // SMALL*SMALL
#define L2     256       // LARGE*LARGE
#define HW     65536     // H*W
#define NPAIR  (WIN*NB)  // 2048 (w,b) pairs
#define NELEM  ((size_t)WIN*NB*S2*L2)  // 33,554,432

typedef __attribute__((ext_vector_type(2))) float v2f;
typedef __attribute__((ext_vector_type(8))) float v8f;

// ---------------------------------------------------------------------------
// Kernel 1a: zero the partner table (2 x 65536 ints) — scatter target.
__global__ void zero_partner_kernel(int* __restrict__ partner) {
    int i = blockIdx.x * blockDim.x + threadIdx.x;   // grid covers 2*HW
    partner[i] = 0;
}

// Kernel 1b: partner[b, idx1[b,j]] = idx0[b,j]
// index_r layout: (B, 2, HW) int32, row-major.
__global__ void scatter_partner_kernel(const int* __restrict__ index_r,
                                       int* __restrict__ partner) {
    int i = blockIdx.x * blockDim.x + threadIdx.x;   // 0 .. 2*HW-1
    int b = i >> 16;
    int j = i & (HW - 1);
    int idx0 = index_r[(b * 2 + 0) * HW + j];
    int idx1 = index_r[(b * 2 + 1) * HW + j];
    partner[(b << 16) + idx1] = idx0;
}

// ---------------------------------------------------------------------------
// Kernel 2: per-(w,b) pair. Block = 256 threads, thread t owns column l=t.
// P flat index: ((w*NB + b)*S2 + s)*L2 + l  -> base wb*16384, stride 256 in s.
// Since sw_abs[w,:] holds 64 distinct pixel ids, each thread matches at most
// one s. So sum|P-M| = sum(P) + correction at the single match, and log() is
// evaluated only at the match. Main loop is a pure coalesced stream-sum.
__global__ void pair_kernel(const float* __restrict__ P,
                            const int*  __restrict__ partner,
                            const int*  __restrict__ sw_abs,
                            const int*  __restrict__ lw_abs,
                            float* __restrict__ pair_nll,   // [NPAIR] : nll/cnt
                            float* __restrict__ pair_abs) { // [NPAIR] : sum|P-M|
    const int wb = blockIdx.x;           // 0..2047
    const int w  = wb >> 1;
    const int b  = wb & 1;
    const int t  = threadIdx.x;          // 0..255 == l

    __shared__ int   sw_s[S2];
    __shared__ float wsum[3][8];

    if (t < S2) sw_s[t] = sw_abs[(w << 6) + t];
    const int lw = lw_abs[(w << 8) + t];
    const int pl = partner[(b << 16) + lw];
    __syncthreads();

    // Find the unique matching s (or -1). zmask kills sw==0 && lw==0; since a
    // match implies sw==pl, that is pl==0 && lw==0.
    int ms = -1;
    const bool killed = (pl == 0) && (lw == 0);
    #pragma unroll
    for (int s = 0; s < S2; ++s)
        if (sw_s[s] == pl) ms = s;
    if (killed) ms = -1;

    const float* Pp = P + ((size_t)wb << 14) + t;

    // Stream-sum of this thread's 64 elements (memory-bound inner loop).
    float psum = 0.0f;
    #pragma unroll 8
    for (int s = 0; s < S2; ++s)
        psum += Pp[(size_t)(s << 8)];

    float cnt = 0.0f, nll = 0.0f, absd = psum;
    if (ms >= 0) {
        float pm = Pp[(size_t)(ms << 8)];
        cnt = 1.0f;
        float pc = fminf(fmaxf(pm, 1e-6f), 1.0f - 1e-6f);
        nll = -__logf(pc);
        absd += fabsf(pm - 1.0f) - fabsf(pm);  // replace |p-0| by |p-1|
    }

    // Wave32 shfl reduction, then across the 8 waves via LDS.
    #pragma unroll
    for (int off = 16; off > 0; off >>= 1) {
        cnt  += __shfl_xor(cnt,  off);
        nll  += __shfl_xor(nll,  off);
        absd += __shfl_xor(absd, off);
    }
    const int wave = t >> 5;
    if ((t & 31) == 0) {
        wsum[0][wave] = cnt; wsum[1][wave] = nll; wsum[2][wave] = absd;
    }
    __syncthreads();
    if (t == 0) {
        float c = 0.f, n = 0.f, a = 0.f;
        #pragma unroll
        for (int i = 0; i < 8; ++i) { c += wsum[0][i]; n += wsum[1][i]; a += wsum[2][i]; }
        pair_nll[wb] = n / c;     // per-(w,b): -sum(log*M)/c_num
        pair_abs[wb] = a;         // per-(w,b): sum|P-M|
    }
}

// ---------------------------------------------------------------------------
// Kernel 3: single-wave WMMA reduction of the two 2048-entry partial arrays.
// D = A x B + C with B = all-ones gives per-row sums accumulating in C:
//   D[m][j] = sum_k A[m][k] + C[m][j]  (all columns identical).
// A 16x4 f32 layout: lanes 0-15 hold rows m, VGPR0=K0,VGPR1=K1; lanes 16-31
// hold K2,K3 of the same rows -> chunk element index = m*4 + k.
// C/D layout: lane j (0-15) VGPR r = D[r][j]; lane 16+j VGPR r = D[8+r][j].
__global__ void final_reduce_kernel(const float* __restrict__ pair_nll,
                                    const float* __restrict__ pair_abs,
                                    float* __restrict__ out) {
    const int lane = threadIdx.x;          // 0..31, EXEC all ones
    const int m  = lane & 15;
    const int kb = (lane >> 4) << 1;       // 0 or 2

    v2f ones; ones.x = 1.0f; ones.y = 1.0f;
    v8f accA = {}; v8f accB = {};

    for (int c = 0; c < NPAIR; c += 64) {
        v2f a, bb;
        a.x  = pair_nll[c + m * 4 + kb];
        a.y  = pair_nll[c + m * 4 + kb + 1];
        bb.x = pair_abs[c + m * 4 + kb];
        bb.y = pair_abs[c + m * 4 + kb + 1];
        accA = __builtin_amdgcn_wmma_f32_16x16x4_f32(
            false, a,  false, ones, (short)0, accA, false, false);
        accB = __builtin_amdgcn_wmma_f32_16x16x4_f32(
            false, bb, false, ones, (short)0, accB, false, false);
    }

    // Column 0 lives in lane 0 (rows 0-7) and lane 16 (rows 8-15).
    float sA = accA[0]+accA[1]+accA[2]+accA[3]+accA[4]+accA[5]+accA[6]+accA[7];
    float sB = accB[0]+accB[1]+accB[2]+accB[3]+accB[4]+accB[5]+accB[6]+accB[7];

    __shared__ float tA[32], tB[32];
    tA[lane] = sA; tB[lane] = sB;
    __syncthreads();
    if (lane == 0) {
        float totA = tA[0] + tA[16];
        float totB = tB[0] + tB[16];
        out[0] = totA / (float)NPAIR;      // l_cm
        out[1] = totB / (float)NELEM;      // l_c
    }
}

// ---------------------------------------------------------------------------
extern "C" void kernel_launch(void* const* d_in, const int* in_sizes, int n_in,
                              void* d_out, int out_size, void* d_ws, size_t ws_size,
                              hipStream_t stream) {
    const float* P       = (const float*)d_in[0]; // (1024,2,64,256) f32
    const int*   index_r = (const int*)  d_in[1]; // (2,2,65536) i32
    const int*   sw_abs  = (const int*)  d_in[2]; // (1024,64) i32
    const int*   lw_abs  = (const int*)  d_in[3]; // (1024,256) i32

    // Workspace layout: partner table (2*HW ints), then two partial arrays.
    int*   partner  = (int*)d_ws;
    float* pair_nll = (float*)((char*)d_ws + (size_t)2 * HW * sizeof(int));
    float* pair_abs = pair_nll + NPAIR;
    float* out      = (float*)d_out;

    zero_partner_kernel   <<<(2 * HW) / 256, 256, 0, stream>>>(partner);
    scatter_partner_kernel<<<(2 * HW) / 256, 256, 0, stream>>>(index_r, partner);
    pair_kernel           <<<NPAIR, 256, 0, stream>>>(P, partner, sw_abs, lw_abs,
                                                      pair_nll, pair_abs);
    final_reduce_kernel   <<<1, 32, 0, stream>>>(pair_nll, pair_abs, out);
}